// Encoder_42013370089615
// MI455X (gfx1250) — compile-verified
//
#include <hip/hip_runtime.h>
#include <math.h>

#define NB   4096
#define ND   784
#define ND1  392
#define NH   16
#define NL   4
#define NR   4
#define NK   8
#define PCOL (ND1 * 3 * NK)   // 9408
#define NBIS 50
#define NNEWT 3
#define F_LOG2PI 1.8378770664093453f

typedef __attribute__((ext_vector_type(2))) float v2f;
typedef __attribute__((ext_vector_type(8))) float v8f;

__device__ __forceinline__ v8f wmma4(v2f a, v2f b, v8f c) {
  // V_WMMA_F32_16X16X4_F32 : D = A(16x4) * B(4x16) + C(16x16)
  return __builtin_amdgcn_wmma_f32_16x16x4_f32(false, a, false, b, (short)0, c,
                                               false, false);
}

// Fast transcendentals: lower to v_exp_f32 / v_log_f32. All call sites here
// have range-safe arguments (see analysis), so no ocml range reduction needed.
__device__ __forceinline__ float fexp(float x) { return __expf(x); }
__device__ __forceinline__ float flog(float x) { return __logf(x); }

__device__ __forceinline__ float lse8(const float* v) {
  float mx = v[0];
#pragma unroll
  for (int k = 1; k < NK; ++k) mx = fmaxf(mx, v[k]);
  float s = 0.f;
#pragma unroll
  for (int k = 0; k < NK; ++k) s += fexp(v[k] - mx);
  return mx + flog(s);
}

// f = log_cdf - log_sf   (bisection objective, no pdf needed)
// t_k = fma(xv, inv_s_k, nms_k) with nms_k = -mu_k * inv_s_k
__device__ __forceinline__ float mix_f(float xv, const float* logw,
                                       const float* nms, const float* inv_s,
                                       float& lc, float& lsf) {
  float a[NK], bq[NK];
#pragma unroll
  for (int k = 0; k < NK; ++k) {
    float t    = fmaf(xv, inv_s[k], nms[k]);
    float lsig = fminf(t, 0.f) - flog(1.f + fexp(-fabsf(t)));  // log sigmoid(t)
    a[k]  = logw[k] + lsig;          // -> log_cdf terms
    bq[k] = a[k] - t;                // log sigmoid(-t) = lsig - t
  }
  lc  = lse8(a);
  lsf = lse8(bq);
  return lc - lsf;
}

__device__ __forceinline__ void mix_full(float xv, const float* logw,
                                         const float* nms, const float* inv_s,
                                         const float* ls, float& lc, float& lsf,
                                         float& lp) {
  float a[NK], bq[NK], pq[NK];
#pragma unroll
  for (int k = 0; k < NK; ++k) {
    float t    = fmaf(xv, inv_s[k], nms[k]);
    float lsig = fminf(t, 0.f) - flog(1.f + fexp(-fabsf(t)));
    a[k]  = logw[k] + lsig;
    bq[k] = a[k] - t;
    pq[k] = a[k] + (lsig - t) - ls[k];  // logw + lsig(t) + lsig(-t) - log_s
  }
  lc  = lse8(a);
  lsf = lse8(bq);
  lp  = lse8(pq);
}

// ---------------------------------------------------------------------------
// log_q init + z = eps   (one block per batch row, deterministic tree reduce)
// ---------------------------------------------------------------------------
__global__ __launch_bounds__(256) void init_kernel(const float* __restrict__ eps,
                                                   float* __restrict__ z,
                                                   float* __restrict__ logq) {
  __shared__ float red[256];
  const int b = blockIdx.x, t = threadIdx.x;
  float s = 0.f;
  for (int k = t; k < ND; k += 256) {
    float e = eps[b * ND + k];
    z[b * ND + k] = e;
    s += e * e;
  }
  red[t] = s;
  __syncthreads();
  for (int off = 128; off > 0; off >>= 1) {
    if (t < off) red[t] += red[t + off];
    __syncthreads();
  }
  if (t == 0) logq[b] = -0.5f * (red[0] + (float)ND * F_LOG2PI);
}

// ---------------------------------------------------------------------------
// Conditioner hidden state:
//   h = swish(concat(zb,x) @ W_in + b_in); R residual blocks of 16x16 GEMMs.
// 256 threads = 8 waves, each wave owns a 16-row tile (128 rows / block).
// All GEMMs via V_WMMA_F32_16X16X4_F32.
// ---------------------------------------------------------------------------
__global__ __launch_bounds__(256) void hidden_kernel(
    const float* __restrict__ z, const float* __restrict__ x,
    const float* __restrict__ Win, const float* __restrict__ bin,
    const float* __restrict__ W1, const float* __restrict__ b1,
    const float* __restrict__ W2, const float* __restrict__ b2,
    float* __restrict__ h, int zbOff) {
  __shared__ float ldsH[8 * 256];
  __shared__ float ldsU[8 * 256];

  const int lane = threadIdx.x & 31;
  const int wave = threadIdx.x >> 5;
  const int rowBase = blockIdx.x * 128 + wave * 16;
  const int m    = lane & 15;            // A-row / B-col / C-col owned by lane
  const int koff = (lane < 16) ? 0 : 2;  // K sub-offset (ISA 16x4 A layout)
  const int rOff = (lane < 16) ? 0 : 8;  // C rows: vgpr i -> row i + rOff
  const int row  = rowBase + m;

  v8f c = {0.f, 0.f, 0.f, 0.f, 0.f, 0.f, 0.f, 0.f};

  // K = 0..ND1 : zb part of concat input
  for (int k0 = 0; k0 < ND1; k0 += 4) {
    v2f a, b;
    a.x = z[row * ND + zbOff + k0 + koff];
    a.y = z[row * ND + zbOff + k0 + koff + 1];
    b.x = Win[(k0 + koff) * NH + m];
    b.y = Win[(k0 + koff + 1) * NH + m];
    c = wmma4(a, b, c);
  }
  // K = ND1..ND1+ND : x part of concat input
  for (int k0 = ND1; k0 < ND1 + ND; k0 += 4) {
    v2f a, b;
    a.x = x[row * ND + (k0 - ND1) + koff];
    a.y = x[row * ND + (k0 - ND1) + koff + 1];
    b.x = Win[(k0 + koff) * NH + m];
    b.y = Win[(k0 + koff + 1) * NH + m];
    c = wmma4(a, b, c);
  }
  {
    const float bias = bin[m];
#pragma unroll
    for (int i = 0; i < 8; ++i) {
      float v = c[i] + bias;
      c[i] = v / (1.f + fexp(-v));  // swish
    }
  }

  float* myH = ldsH + wave * 256;
  float* myU = ldsU + wave * 256;
#pragma unroll
  for (int i = 0; i < 8; ++i) myH[(i + rOff) * 16 + m] = c[i];
  __syncthreads();

  for (int r = 0; r < NR; ++r) {
    const float* W1r = W1 + r * NH * NH;
    const float* W2r = W2 + r * NH * NH;
    v8f t = {0.f, 0.f, 0.f, 0.f, 0.f, 0.f, 0.f, 0.f};
#pragma unroll
    for (int k0 = 0; k0 < NH; k0 += 4) {
      v2f a, b;
      a.x = myH[m * 16 + k0 + koff];
      a.y = myH[m * 16 + k0 + koff + 1];
      b.x = W1r[(k0 + koff) * NH + m];
      b.y = W1r[(k0 + koff + 1) * NH + m];
      t = wmma4(a, b, t);
    }
    const float b1v = b1[r * NH + m];
#pragma unroll
    for (int i = 0; i < 8; ++i) {
      float v = t[i] + b1v;
      t[i] = v / (1.f + fexp(-v));
    }
#pragma unroll
    for (int i = 0; i < 8; ++i) myU[(i + rOff) * 16 + m] = t[i];
    __syncthreads();

    v8f u = {0.f, 0.f, 0.f, 0.f, 0.f, 0.f, 0.f, 0.f};
#pragma unroll
    for (int k0 = 0; k0 < NH; k0 += 4) {
      v2f a, b;
      a.x = myU[m * 16 + k0 + koff];
      a.y = myU[m * 16 + k0 + koff + 1];
      b.x = W2r[(k0 + koff) * NH + m];
      b.y = W2r[(k0 + koff + 1) * NH + m];
      u = wmma4(a, b, u);
    }
    const float b2v = b2[r * NH + m];
#pragma unroll
    for (int i = 0; i < 8; ++i) c[i] += u[i] + b2v;
#pragma unroll
    for (int i = 0; i < 8; ++i) myH[(i + rOff) * 16 + m] = c[i];
    __syncthreads();
  }

#pragma unroll
  for (int i = 0; i < 8; ++i) h[(rowBase + i + rOff) * NH + m] = c[i];
}

// ---------------------------------------------------------------------------
// Fused output-projection (WMMA) + mixture-logit inversion.
// Block = 224 threads (7 waves). Tile: 16 batch rows x 14 mixture elems
//   = 336 p-columns = 21 WMMA col-tiles (3 per wave), K=16 (4 WMMA steps).
// Phase 2: each thread inverts one (b, j) element from LDS params.
// ---------------------------------------------------------------------------
__global__ __launch_bounds__(224) void invert_kernel(
    float* __restrict__ z, const float* __restrict__ h,
    const float* __restrict__ Wout, const float* __restrict__ bout,
    float* __restrict__ part, int zaOff) {
  __shared__ float ldsP[16 * 336];  // 21.5 KB p-tile
  __shared__ float ldLd[224];

  const int lane = threadIdx.x & 31;
  const int wave = threadIdx.x >> 5;
  const int bTile = blockIdx.x, jTile = blockIdx.y;
  const int rowBase = bTile * 16;
  const int m    = lane & 15;
  const int koff = (lane < 16) ? 0 : 2;
  const int rOff = (lane < 16) ? 0 : 8;

  // A operand (h rows) is identical for all 3 column tiles of this wave:
  // load the 4 K-step fragments once.
  v2f afrag[4];
#pragma unroll
  for (int ks = 0; ks < 4; ++ks) {
    afrag[ks].x = h[(rowBase + m) * NH + ks * 4 + koff];
    afrag[ks].y = h[(rowBase + m) * NH + ks * 4 + koff + 1];
  }

  // Phase 1: p-tile = h(16x16) @ Wout(16x336 slice) + b_out, via WMMA
  for (int ctw = 0; ctw < 3; ++ctw) {
    const int ct   = wave * 3 + ctw;       // 0..20
    const int colB = ct * 16;
    const int gcol = jTile * 336 + colB + m;
    v8f c = {0.f, 0.f, 0.f, 0.f, 0.f, 0.f, 0.f, 0.f};
#pragma unroll
    for (int ks = 0; ks < 4; ++ks) {
      v2f b;
      b.x = Wout[(ks * 4 + koff) * PCOL + gcol];
      b.y = Wout[(ks * 4 + koff + 1) * PCOL + gcol];
      c = wmma4(afrag[ks], b, c);
    }
    const float bias = bout[gcol];
#pragma unroll
    for (int i = 0; i < 8; ++i) ldsP[(i + rOff) * 336 + colB + m] = c[i] + bias;
  }
  __syncthreads();

  // Phase 2: per-element inversion (bisection + Newton)
  const int t  = threadIdx.x;
  const int mr = t / 14, j = t % 14;
  const int b  = rowBase + mr;
  const int jG = jTile * 14 + j;
  const float* pp = ldsP + mr * 336 + j * 24;

  float logw[NK], nms[NK], ls[NK], inv_s[NK];
  {
    float lmx = pp[0];
#pragma unroll
    for (int k = 1; k < NK; ++k) lmx = fmaxf(lmx, pp[k]);
    float se = 0.f;
#pragma unroll
    for (int k = 0; k < NK; ++k) se += fexp(pp[k] - lmx);
    const float lse = lmx + flog(se);
#pragma unroll
    for (int k = 0; k < NK; ++k) {
      logw[k]  = pp[k] - lse;
      ls[k]    = pp[2 * NK + k];
      inv_s[k] = fexp(-ls[k]);
      nms[k]   = -pp[NK + k] * inv_s[k];  // -mu * inv_s (FMA form)
    }
  }

  const float y = z[b * ND + zaOff + jG];
  float lo = -100.f, hi = 100.f, lc, lsf, lp;
#pragma unroll 1
  for (int it = 0; it < NBIS; ++it) {
    const float mid = 0.5f * (lo + hi);
    const float f = mix_f(mid, logw, nms, inv_s, lc, lsf) - y;
    lo = (f < 0.f) ? mid : lo;
    hi = (f < 0.f) ? hi : mid;
  }
  float xv = 0.5f * (lo + hi);
#pragma unroll 1
  for (int it = 0; it < NNEWT; ++it) {
    mix_full(xv, logw, nms, inv_s, ls, lc, lsf, lp);
    xv -= (lc - lsf - y) * fexp(lc + lsf - lp);  // f * exp(-logdet)
  }
  mix_full(xv, logw, nms, inv_s, ls, lc, lsf, lp);
  const float ld = lp - lc - lsf;

  z[b * ND + zaOff + jG] = xv;  // in-place: this thread owns this element
  ldLd[t] = ld;
  __syncthreads();
  if (t < 16) {  // deterministic fixed-order partial sum per batch row
    float s = 0.f;
    for (int jj = 0; jj < 14; ++jj) s += ldLd[t * 14 + jj];
    part[(rowBase + t) * 28 + jTile] = s;
  }
}

__global__ __launch_bounds__(256) void logq_reduce(const float* __restrict__ part,
                                                   float* __restrict__ logq) {
  const int b = blockIdx.x * 256 + threadIdx.x;
  float s = 0.f;
  for (int tt = 0; tt < 28; ++tt) s += part[b * 28 + tt];
  logq[b] += s;
}

extern "C" void kernel_launch(void* const* d_in, const int* in_sizes, int n_in,
                              void* d_out, int out_size, void* d_ws,
                              size_t ws_size, hipStream_t stream) {
  const float* x     = (const float*)d_in[0];
  const float* eps   = (const float*)d_in[1];
  const float* W_in  = (const float*)d_in[2];
  const float* b_in  = (const float*)d_in[3];
  const float* W1    = (const float*)d_in[4];
  const float* b1    = (const float*)d_in[5];
  const float* W2    = (const float*)d_in[6];
  const float* b2    = (const float*)d_in[7];
  const float* W_out = (const float*)d_in[8];
  const float* b_out = (const float*)d_in[9];

  float* z    = (float*)d_out;                       // B x D, updated in place
  float* logq = (float*)d_out + (size_t)NB * ND;     // B
  float* h    = (float*)d_ws;                        // B x H
  float* part = (float*)d_ws + (size_t)NB * NH;      // B x 28 log-det partials

  init_kernel<<<NB, 256, 0, stream>>>(eps, z, logq);

  for (int l = NL - 1; l >= 0; --l) {
    const int zbOff = (l & 1) ? 0 : ND1;   // zb half of z
    const int zaOff = (l & 1) ? ND1 : 0;   // za half (read y, write xa)
    hidden_kernel<<<NB / 128, 256, 0, stream>>>(
        z, x, W_in + (size_t)l * (ND1 + ND) * NH, b_in + l * NH,
        W1 + (size_t)l * NR * NH * NH, b1 + l * NR * NH,
        W2 + (size_t)l * NR * NH * NH, b2 + l * NR * NH, h, zbOff);
    dim3 g(NB / 16, ND1 / 14);
    invert_kernel<<<g, 224, 0, stream>>>(
        z, h, W_out + (size_t)l * NH * PCOL, b_out + (size_t)l * PCOL, part,
        zaOff);
    logq_reduce<<<NB / 256, 256, 0, stream>>>(part, logq);
  }
}